// TrustRegionBase_20744692040052
// MI455X (gfx1250) — compile-verified
//
#include <hip/hip_runtime.h>

// ---------------------------------------------------------------------------
// Dense photometric trust-region alignment (CDNA5 / gfx1250, wave32).
//  - weights kernel: residual at initial pose -> w = 1/(1+r^2) (output)
//  - G kernel:       G = sum w * g g^T  with g = [J(6) | r], via
//                    V_WMMA_F32_16X16X4_F32 outer-product accumulation
//  - solve kernel:   (JtWJ + 1e-6 I) xi = JtR ; pose <- se3_exp(-xi) * pose
// Everything fused & recomputed; scratch = pose + 7x7 accumulators only.
// ---------------------------------------------------------------------------

typedef float v2f __attribute__((ext_vector_type(2)));
typedef float v8f __attribute__((ext_vector_type(8)));

#define B_N   8
#define C_N   16
#define H_N   240
#define W_N   320
#define HW_N  (H_N * W_N)          // 76800
#define S_N   (C_N * HW_N)         // 1228800 samples per batch
#define OCC_T 0.1f
#define DAMP  1e-6f

#define GBLK  320                  // blocks per batch for k_G
#define GTHR  128                  // threads per block (4 waves)
#define GITER (S_N / (GBLK * GTHR))// = 30, exact

__device__ __forceinline__ float bilin2(const float* __restrict__ pl,
                                        int u0, int v0, int u1, int v1,
                                        float wu, float wv)
{
    float f00 = pl[v0 * W_N + u0];
    float f01 = pl[v0 * W_N + u1];
    float f10 = pl[v1 * W_N + u0];
    float f11 = pl[v1 * W_N + u1];
    return f00 * (1.f - wu) * (1.f - wv) + f01 * wu * (1.f - wv)
         + f10 * (1.f - wu) * wv         + f11 * wu * wv;
}

// Warped residual for sample (b,c,h,w); also returns px,py,invD0 for Jacobian.
__device__ __forceinline__ float residual_px(
    int b, int c, int h, int w,
    const float* Rm, const float* tv,
    float fx, float fy, float cx, float cy,
    const float* __restrict__ x0, const float* __restrict__ x1,
    const float* __restrict__ invD0, const float* __restrict__ invD1,
    float& px_o, float& py_o, float& id0_o)
{
    float px = ((float)w - cx) / fx;
    float py = ((float)h - cy) / fy;
    int   p  = h * W_N + w;
    float id0 = invD0[(size_t)b * HW_N + p];

    float X = Rm[0]*px + Rm[1]*py + Rm[2] + tv[0]*id0;
    float Y = Rm[3]*px + Rm[4]*py + Rm[5] + tv[1]*id0;
    float Z = Rm[6]*px + Rm[7]*py + Rm[8] + tv[2]*id0;
    float u = X / Z * fx + cx;
    float v = Y / Z * fy + cy;
    float inv_z = id0 / Z;

    float uc = fminf(fmaxf(u, 0.f), (float)(W_N - 1));
    float vc = fminf(fmaxf(v, 0.f), (float)(H_N - 1));
    float u0f = floorf(uc), v0f = floorf(vc);
    float wu = uc - u0f, wvv = vc - v0f;
    int u0 = (int)u0f, v0 = (int)v0f;
    int u1 = min(u0 + 1, W_N - 1), v1 = min(v0 + 1, H_N - 1);

    float idz = bilin2(invD1 + (size_t)b * HW_N, u0, v0, u1, v1, wu, wvv);
    bool inlier = inv_z > (idz - OCC_T);
    bool inview = inlier && (u > 0.f) && (u < (float)(W_N - 1))
                         && (v > 0.f) && (v < (float)(H_N - 1));

    const float* x1c = x1 + ((size_t)b * C_N + c) * HW_N;
    float x1w = bilin2(x1c, u0, v0, u1, v1, wu, wvv);
    float x0v = x0[((size_t)b * C_N + c) * HW_N + p];
    float r = inview ? (x1w - x0v) : 0.001f;

    px_o = px; py_o = py; id0_o = id0;
    return r;
}

// ---------------- initial residual -> weights (output) ---------------------
__global__ void k_weights(const float* __restrict__ poseR, const float* __restrict__ poseT,
                          const float* __restrict__ x0, const float* __restrict__ x1,
                          const float* __restrict__ invD0, const float* __restrict__ invD1,
                          const float* __restrict__ Kmat, float* __restrict__ w_out)
{
    int idx = blockIdx.x * blockDim.x + threadIdx.x;
    if (idx >= B_N * S_N) return;
    int b = idx / S_N;
    int s = idx - b * S_N;
    int c = s / HW_N;
    int p = s - c * HW_N;
    int h = p / W_N;
    int w = p - h * W_N;

    float Rm[9], tv[3];
#pragma unroll
    for (int j = 0; j < 9; ++j) Rm[j] = poseR[b * 9 + j];
#pragma unroll
    for (int j = 0; j < 3; ++j) tv[j] = poseT[b * 3 + j];
    float fx = Kmat[b*4+0], fy = Kmat[b*4+1], cx = Kmat[b*4+2], cy = Kmat[b*4+3];

    float px, py, id0;
    float r = residual_px(b, c, h, w, Rm, tv, fx, fy, cx, cy,
                          x0, x1, invD0, invD1, px, py, id0);
    w_out[idx] = 1.f / (1.f + r * r);
}

// ---------------- G = sum w * g g^T  via WMMA ------------------------------
__global__ void k_G(const float* __restrict__ Rws, const float* __restrict__ tws,
                    const float* __restrict__ x0, const float* __restrict__ x1,
                    const float* __restrict__ invD0, const float* __restrict__ invD1,
                    const float* __restrict__ Kmat, const float* __restrict__ w_in,
                    float* __restrict__ Gacc)
{
    __shared__ float sh_g [4][32][8];  // per-wave staging: g vectors
    __shared__ float sh_wg[4][32][8];  // per-wave staging: w*g vectors

    const int tid  = threadIdx.x;
    const int wv   = tid >> 5;
    const int lane = tid & 31;
    const int b    = blockIdx.y;

    float Rm[9], tv[3];
#pragma unroll
    for (int j = 0; j < 9; ++j) Rm[j] = Rws[b * 9 + j];
#pragma unroll
    for (int j = 0; j < 3; ++j) tv[j] = tws[b * 3 + j];
    const float fx = Kmat[b*4+0], fy = Kmat[b*4+1], cx = Kmat[b*4+2], cy = Kmat[b*4+3];

    v8f acc = {0.f, 0.f, 0.f, 0.f, 0.f, 0.f, 0.f, 0.f};

    const int half = lane >> 4;           // K-half of the 16x16x4 tile
    const int m    = lane & 15;           // M (A) / N (B) index
    const int mi   = (m < 7) ? m : 7;     // LDS row 7 is always zero -> no divergence

    for (int it = 0; it < GITER; ++it) {
        const int s = it * (GBLK * GTHR) + blockIdx.x * GTHR + tid;
        const int c = s / HW_N;
        const int p = s - c * HW_N;
        const int h = p / W_N;
        const int w = p - h * W_N;

        float px, py, id0;
        const float r = residual_px(b, c, h, w, Rm, tv, fx, fy, cx, cy,
                                    x0, x1, invD0, invD1, px, py, id0);

        // edge-clamped Sobel feature gradient of x0 (normalized)
        const float* x0c = x0 + ((size_t)b * C_N + c) * HW_N;
        const int hm = max(h - 1, 0), hp = min(h + 1, H_N - 1);
        const int wm = max(w - 1, 0), wp = min(w + 1, W_N - 1);
        const float a00 = x0c[hm*W_N+wm], a01 = x0c[hm*W_N+w ], a02 = x0c[hm*W_N+wp];
        const float a10 = x0c[h *W_N+wm],                        a12 = x0c[h *W_N+wp];
        const float a20 = x0c[hp*W_N+wm], a21 = x0c[hp*W_N+w ], a22 = x0c[hp*W_N+wp];
        const float dx = (a02 - a00) + 2.f * (a12 - a10) + (a22 - a20);
        const float dy = (a20 - a00) + 2.f * (a21 - a01) + (a22 - a02);
        const float mag = sqrtf(dx * dx + dy * dy + 1e-8f);
        const float gx = dx / mag, gy = dy / mag;

        const float xy = px * py;
        float g[7];
        g[0] = gx * fx * (-xy)           + gy * fy * (-1.f - py * py);
        g[1] = gx * fx * (1.f + px * px) + gy * fy * ( xy);
        g[2] = gx * fx * (-py)           + gy * fy * ( px);
        g[3] = gx * fx * ( id0);
        g[4] =                             gy * fy * ( id0);
        g[5] = gx * fx * (-id0 * px)     + gy * fy * (-id0 * py);
        g[6] = r;
        const float wgt = w_in[(size_t)b * S_N + s];

#pragma unroll
        for (int j = 0; j < 7; ++j) { sh_g[wv][lane][j] = g[j]; sh_wg[wv][lane][j] = wgt * g[j]; }
        sh_g[wv][lane][7] = 0.f; sh_wg[wv][lane][7] = 0.f;
        __syncthreads();

        // 8 chunks of K=4 samples: A[m][k]=w*g_k[m], B[k][n]=g_k[n]
        // -> D += A*B accumulates G[m][n] = sum_k w_k g_k[m] g_k[n]
#pragma unroll
        for (int cc = 0; cc < 8; ++cc) {
            const int s0 = cc * 4 + half * 2;   // this lane-half supplies 2 of 4 K-slots
            v2f Av, Bv;
            Av.x = sh_wg[wv][s0    ][mi];
            Av.y = sh_wg[wv][s0 + 1][mi];
            Bv.x = sh_g [wv][s0    ][mi];
            Bv.y = sh_g [wv][s0 + 1][mi];
            acc = __builtin_amdgcn_wmma_f32_16x16x4_f32(
                      false, Av, false, Bv, (short)0, acc, false, false);
        }
        __syncthreads();
    }

    // D layout: lane n (0..15) holds G[j][n] in acc[j], j = 0..7
    if (lane < 7) {
        float* Gb = Gacc + b * 49;
        float cv[8];
#pragma unroll
        for (int j = 0; j < 8; ++j) cv[j] = acc[j];
#pragma unroll
        for (int j = 0; j < 7; ++j) atomicAdd(&Gb[j * 7 + lane], cv[j]);
    }
}

// ---------------- tiny helpers --------------------------------------------
__global__ void k_zero(float* __restrict__ Gacc)
{
    int i = blockIdx.x * blockDim.x + threadIdx.x;
    if (i < B_N * 49) Gacc[i] = 0.f;
}

__global__ void k_init(const float* __restrict__ Rin, const float* __restrict__ tin,
                       float* __restrict__ Rws, float* __restrict__ tws)
{
    int i = threadIdx.x;
    if (i < B_N * 9) Rws[i] = Rin[i];
    if (i < B_N * 3) tws[i] = tin[i];
}

__global__ void k_final(const float* __restrict__ Rws, const float* __restrict__ tws,
                        float* __restrict__ out)
{
    int i = threadIdx.x;
    if (i < B_N * 9) out[i] = Rws[i];
    if (i < B_N * 3) out[B_N * 9 + i] = tws[i];
}

// ---------------- 6x6 solve + se3_exp + pose compose (1 thread / batch) ----
__global__ void k_solve(const float* __restrict__ Gacc,
                        float* __restrict__ Rws, float* __restrict__ tws)
{
    int b = threadIdx.x;
    if (b >= B_N) return;
    const float* G = Gacc + b * 49;

    float A[6][7];
    for (int i = 0; i < 6; ++i) {
        for (int j = 0; j < 6; ++j) A[i][j] = G[i * 7 + j];
        A[i][i] += DAMP;
        A[i][6] = G[i * 7 + 6];          // JtR = column 6 of G
    }
    // Gauss-Jordan with partial pivoting
    for (int k = 0; k < 6; ++k) {
        int piv = k; float mx = fabsf(A[k][k]);
        for (int r = k + 1; r < 6; ++r) {
            float v = fabsf(A[r][k]);
            if (v > mx) { mx = v; piv = r; }
        }
        if (piv != k)
            for (int j = 0; j < 7; ++j) { float t = A[k][j]; A[k][j] = A[piv][j]; A[piv][j] = t; }
        float d = A[k][k];
        for (int j = k; j < 7; ++j) A[k][j] /= d;
        for (int r = 0; r < 6; ++r) {
            if (r == k) continue;
            float f = A[r][k];
            for (int j = k; j < 7; ++j) A[r][j] -= f * A[k][j];
        }
    }

    float xi[6];
    for (int i = 0; i < 6; ++i) xi[i] = -A[i][6];   // se3_exp(-xi)

    float wx = xi[0], wy = xi[1], wz = xi[2];
    float vx = xi[3], vy = xi[4], vz = xi[5];
    float t2 = wx*wx + wy*wy + wz*wz;
    float th = sqrtf(t2 + 1e-12f);
    bool small = t2 < 1e-8f;
    float Af = small ? (1.f - t2 / 6.f)        : (sinf(th) / th);
    float Bf = small ? (0.5f - t2 / 24.f)      : ((1.f - cosf(th)) / (t2 + 1e-12f));
    float Cf = small ? (1.f/6.f - t2 / 120.f)  : ((1.f - Af) / (t2 + 1e-12f));

    float Wm[9] = {0.f, -wz,  wy,
                   wz,  0.f, -wx,
                  -wy,  wx,  0.f};
    float W2[9];
    for (int i = 0; i < 3; ++i)
        for (int j = 0; j < 3; ++j) {
            float s = 0.f;
            for (int k = 0; k < 3; ++k) s += Wm[i*3+k] * Wm[k*3+j];
            W2[i*3+j] = s;
        }
    float dR[9], Vm[9];
    for (int i = 0; i < 9; ++i) {
        float I = (i % 4 == 0) ? 1.f : 0.f;
        dR[i] = I + Af * Wm[i] + Bf * W2[i];
        Vm[i] = I + Bf * Wm[i] + Cf * W2[i];
    }
    float dt[3];
    for (int i = 0; i < 3; ++i)
        dt[i] = Vm[i*3+0]*vx + Vm[i*3+1]*vy + Vm[i*3+2]*vz;

    float Ro[9], to[3];
    for (int i = 0; i < 9; ++i) Ro[i] = Rws[b*9+i];
    for (int i = 0; i < 3; ++i) to[i] = tws[b*3+i];
    float Rn[9], tn[3];
    for (int i = 0; i < 3; ++i) {
        for (int j = 0; j < 3; ++j) {
            float s = 0.f;
            for (int k = 0; k < 3; ++k) s += dR[i*3+k] * Ro[k*3+j];
            Rn[i*3+j] = s;
        }
        tn[i] = dR[i*3+0]*to[0] + dR[i*3+1]*to[1] + dR[i*3+2]*to[2] + dt[i];
    }
    for (int i = 0; i < 9; ++i) Rws[b*9+i] = Rn[i];
    for (int i = 0; i < 3; ++i) tws[b*3+i] = tn[i];
}

// ---------------------------------------------------------------------------
extern "C" void kernel_launch(void* const* d_in, const int* in_sizes, int n_in,
                              void* d_out, int out_size, void* d_ws, size_t ws_size,
                              hipStream_t stream)
{
    (void)in_sizes; (void)n_in; (void)out_size; (void)ws_size;
    const float* poseR = (const float*)d_in[0];
    const float* poseT = (const float*)d_in[1];
    const float* x0    = (const float*)d_in[2];
    const float* x1    = (const float*)d_in[3];
    const float* invD0 = (const float*)d_in[4];
    const float* invD1 = (const float*)d_in[5];
    const float* Kmat  = (const float*)d_in[6];

    float* out  = (float*)d_out;
    float* Rws  = (float*)d_ws;            // 72 floats
    float* tws  = Rws + B_N * 9;           // 24 floats
    float* Gacc = tws + B_N * 3;           // 392 floats
    float* w_out = out + B_N * 9 + B_N * 3;  // weights output region

    k_init<<<1, 128, 0, stream>>>(poseR, poseT, Rws, tws);
    k_weights<<<(B_N * S_N + 255) / 256, 256, 0, stream>>>(
        poseR, poseT, x0, x1, invD0, invD1, Kmat, w_out);

    for (int it = 0; it < 3; ++it) {
        k_zero<<<2, 256, 0, stream>>>(Gacc);
        dim3 grid(GBLK, B_N);
        k_G<<<grid, GTHR, 0, stream>>>(Rws, tws, x0, x1, invD0, invD1, Kmat, w_out, Gacc);
        k_solve<<<1, 32, 0, stream>>>(Gacc, Rws, tws);
    }
    k_final<<<1, 128, 0, stream>>>(Rws, tws, out);
}